// SecondOrderIntersection_32238024524281
// MI455X (gfx1250) — compile-verified
//
#include <hip/hip_runtime.h>

// WMMA operand vector types (per-lane register footprints)
typedef __attribute__((ext_vector_type(2))) float v2f;  // A/B of 16x16x4 f32: 2 VGPRs
typedef __attribute__((ext_vector_type(4))) float v4f;  // global b128 staging
typedef __attribute__((ext_vector_type(8))) float v8f;  // C/D 16x16 f32: 8 VGPRs

#define FDIM    64
#define DDIM    256
#define STRIDE  260          // padded LDS row stride (floats): bank-conflict-free
#define NPAIR   2016         // 64*63/2
#define THREADS 320          // 10 waves -> one 16x16 output tile each

__global__ __launch_bounds__(THREADS) void
SecondOrderIntersection_kernel(const float* __restrict__ in, float* __restrict__ out)
{
    __shared__ float ldsX[FDIM * STRIDE];   // 64*260*4 = 66,560 B

    const int b   = blockIdx.x;
    const int tid = threadIdx.x;
    const float* __restrict__ src = in + (size_t)b * (FDIM * DDIM);

    // ---- cooperative staged load: 4096 float4, fully coalesced in global ----
    for (int idx = tid; idx < (FDIM * DDIM / 4); idx += THREADS) {
        const int row = idx >> 6;      // 64 float4 per row
        const int c4  = idx & 63;
        const v4f v = *(const v4f*)(src + (row << 8) + (c4 << 2));
        *(v4f*)(&ldsX[row * STRIDE + (c4 << 2)]) = v;
    }
    __syncthreads();

    // ---- tile assignment: 10 tiles (ti<=tj) of the 4x4 tile grid ----
    static const unsigned char TI[10] = {0,0,0,0,1,1,1,2,2,3};
    static const unsigned char TJ[10] = {0,1,2,3,1,2,3,2,3,3};
    const int wave = tid >> 5;
    const int ti = TI[wave];
    const int tj = TJ[wave];

    const int lane = tid & 31;
    const int lr   = lane & 15;       // row within tile / column index
    const int hi   = lane >> 4;       // half-wave select

    // WMMA f32 16x16x4 operand layout: lane(lr,hi), VGPR v holds element
    // [row0+lr][k + v + 2*hi]  -> contiguous float pair at (k + 2*hi).
    const float* __restrict__ arow = &ldsX[(ti * 16 + lr) * STRIDE + (hi << 1)];
    const float* __restrict__ brow = &ldsX[(tj * 16 + lr) * STRIDE + (hi << 1)];

    v8f acc = {0.f, 0.f, 0.f, 0.f, 0.f, 0.f, 0.f, 0.f};

    #pragma unroll 8
    for (int k = 0; k < DDIM; k += 4) {
        const v2f a = *(const v2f*)(arow + k);
        const v2f bb = *(const v2f*)(brow + k);
        // D = A(16x4) * B(4x16) + C ; exact fp32 matrix op
        acc = __builtin_amdgcn_wmma_f32_16x16x4_f32(
            /*neg_a=*/false, a, /*neg_b=*/false, bb,
            /*c_mod=*/(short)0, acc, /*reuse_a=*/false, /*reuse_b=*/false);
    }

    // ---- scatter strict upper triangle (i<j), triu_indices(64,k=1) order ----
    float* __restrict__ outb = out + (size_t)b * NPAIR;
    #pragma unroll
    for (int v = 0; v < 8; ++v) {
        const int i = ti * 16 + v + (hi << 3);   // C-matrix row for this VGPR/lane
        const int j = tj * 16 + lr;              // C-matrix col
        if (i < j) {
            const int p = i * 63 - (i * (i - 1)) / 2 + (j - i - 1);
            outb[p] = acc[v];
        }
    }
}

extern "C" void kernel_launch(void* const* d_in, const int* in_sizes, int n_in,
                              void* d_out, int out_size, void* d_ws, size_t ws_size,
                              hipStream_t stream) {
    const float* in  = (const float*)d_in[0];
    float*       out = (float*)d_out;
    const int nbatch = in_sizes[0] / (FDIM * DDIM);   // 4096
    hipLaunchKernelGGL(SecondOrderIntersection_kernel,
                       dim3(nbatch), dim3(THREADS), 0, stream, in, out);
}